// CRF_755914244160
// MI455X (gfx1250) — compile-verified
//
#include <hip/hip_runtime.h>
#include <math.h>
#include <stdint.h>

// ---------------------------------------------------------------------------
// CRF forward-backward, chunked parallel scan for MI455X (gfx1250, wave32).
//
// ws layout (floats):
//   Pf : [B][NC][32][32]   composed per-chunk log-matrices   (8.5 MB)
//   Aw : [B][NC+1][32]     alpha at chunk boundaries
//   Bw : [B][NC+1][32]     beta  at chunk boundaries
//   LA : [B][4098][32]     full log-alpha                    (16.8 MB)
// total ~25.8 MB
// ---------------------------------------------------------------------------

#define NEGV (-1.0e9f)

typedef __attribute__((ext_vector_type(2))) float v2f;
typedef __attribute__((ext_vector_type(8))) float v8f;

static constexpr int B_ = 32, Y_ = 32, T1 = 4097, NODES = 4098, CH = 64, NC = 65;

static constexpr size_t PF_OFF = 0;
static constexpr size_t PF_SZ  = (size_t)B_ * NC * Y_ * Y_;
static constexpr size_t AW_OFF = PF_OFF + PF_SZ;
static constexpr size_t AW_SZ  = (size_t)B_ * (NC + 1) * Y_;
static constexpr size_t BW_OFF = AW_OFF + AW_SZ;
static constexpr size_t BW_SZ  = AW_SZ;
static constexpr size_t LA_OFF = BW_OFF + BW_SZ;

// Low 32 bits of a flat pointer into LDS == LDS byte offset (aperture in high bits).
__device__ __forceinline__ uint32_t lds_off(const void* p) {
  return (uint32_t)(uintptr_t)p;
}

// Per-lane async copy of 16 bytes: from global base saddr plus offset voff to the LDS address in ldsa (GVS mode).
__device__ __forceinline__ void async_b128(uint32_t ldsa, uint32_t voff,
                                           const float* saddr) {
  asm volatile("global_load_async_to_lds_b128 %0, %1, %2"
               :: "v"(ldsa), "v"(voff), "s"(saddr) : "memory");
}
__device__ __forceinline__ void wait_async0() {
  asm volatile("s_wait_asynccnt 0x0" ::: "memory");
}

// ---------------------------------------------------------------------------
// Kernel A: per-(b,chunk) wave composes chunk's matrices in linear space with
// a running log scale, using V_WMMA_F32_16X16X4_F32 (32 WMMA per compose).
// Next matrix is prefetched into a ping-pong LDS buffer with async-to-LDS.
// ---------------------------------------------------------------------------
__global__ __launch_bounds__(32) void crf_compose(const float* __restrict__ lm,
                                                  float* __restrict__ ws) {
  const int c = blockIdx.x, b = blockIdx.y;
  const int lane = threadIdx.x;
  const int L = min(CH, T1 - c * CH);

  __shared__ float Pb[32][33];               // running product, pitch 33 (odd)
  __shared__ float Eb[32][33];               // exp of current transition matrix
  __shared__ __align__(16) float Mraw[2][1024];  // async ping-pong staging

  const float* mbase = lm + ((size_t)b * T1 + (size_t)c * CH) * 1024;

  for (int r = 0; r < 32; ++r) Pb[r][lane] = (r == lane) ? 1.0f : 0.0f;
  __syncthreads();

  const int half = lane >> 4;
  const int lr   = lane & 15;

  // prologue: kick off async load of matrix s=0 into buffer 0
  {
    const uint32_t gv = (uint32_t)(lane * 16);
#pragma unroll
    for (int k = 0; k < 8; ++k)
      async_b128(lds_off(&Mraw[0][k * 128 + lane * 4]), gv + k * 512, mbase);
  }

  float gscale = 0.0f, inv = 1.0f;
  v8f cc00 = {0,0,0,0,0,0,0,0}, cc01 = cc00, cc10 = cc00, cc11 = cc00;

  for (int s = 0; s < L; ++s) {
    const int cur = s & 1;
    wait_async0();                            // Mraw[cur] ready

    if (s + 1 < L) {                          // prefetch next matrix
      const uint32_t gv = (uint32_t)((s + 1) * 4096 + lane * 16);
#pragma unroll
      for (int k = 0; k < 8; ++k)
        async_b128(lds_off(&Mraw[1 - cur][k * 128 + lane * 4]), gv + k * 512, mbase);
    }

    // E = exp(m_t) from staged LDS
    for (int r = 0; r < 32; ++r)
      Eb[r][lane] = __expf(Mraw[cur][r * 32 + lane]);
    __syncthreads();

    // P_new = P_old x E  via 2x2 tiles of V_WMMA_F32_16X16X4_F32
    v8f z = {0,0,0,0,0,0,0,0};
    cc00 = z; cc01 = z; cc10 = z; cc11 = z;
#pragma unroll
    for (int kb = 0; kb < 32; kb += 4) {
      const int k0 = kb + 2 * half;           // lane half selects K pair
      v2f a0, a1, b0, b1;
      a0.x = Pb[lr][k0];       a0.y = Pb[lr][k0 + 1];
      a1.x = Pb[16 + lr][k0];  a1.y = Pb[16 + lr][k0 + 1];
      b0.x = Eb[k0][lr];       b0.y = Eb[k0 + 1][lr];
      b1.x = Eb[k0][16 + lr];  b1.y = Eb[k0 + 1][16 + lr];
      cc00 = __builtin_amdgcn_wmma_f32_16x16x4_f32(false, a0, false, b0, (short)0, cc00, false, false);
      cc01 = __builtin_amdgcn_wmma_f32_16x16x4_f32(false, a0, false, b1, (short)0, cc01, false, false);
      cc10 = __builtin_amdgcn_wmma_f32_16x16x4_f32(false, a1, false, b0, (short)0, cc10, false, false);
      cc11 = __builtin_amdgcn_wmma_f32_16x16x4_f32(false, a1, false, b1, (short)0, cc11, false, false);
    }

    // renormalize by global max (entries strictly positive)
    float mloc = 0.0f;
#pragma unroll
    for (int r = 0; r < 8; ++r)
      mloc = fmaxf(mloc, fmaxf(fmaxf(cc00[r], cc01[r]), fmaxf(cc10[r], cc11[r])));
#pragma unroll
    for (int off = 16; off > 0; off >>= 1)
      mloc = fmaxf(mloc, __shfl_xor(mloc, off, 32));
    gscale += __logf(mloc);
    inv = 1.0f / mloc;

    __syncthreads();
#pragma unroll
    for (int r = 0; r < 8; ++r) {             // write back normalized product
      const int row = r + 8 * half;
      Pb[row][lr]           = cc00[r] * inv;
      Pb[row][16 + lr]      = cc01[r] * inv;
      Pb[16 + row][lr]      = cc10[r] * inv;
      Pb[16 + row][16 + lr] = cc11[r] * inv;
    }
    __syncthreads();
  }

  float* pf = ws + PF_OFF + ((size_t)b * NC + c) * 1024;
#pragma unroll
  for (int r = 0; r < 8; ++r) {
    const int row = r + 8 * half;
    pf[row * 32 + lr]             = __logf(fmaxf(cc00[r] * inv, 1e-37f)) + gscale;
    pf[row * 32 + 16 + lr]        = __logf(fmaxf(cc01[r] * inv, 1e-37f)) + gscale;
    pf[(16 + row) * 32 + lr]      = __logf(fmaxf(cc10[r] * inv, 1e-37f)) + gscale;
    pf[(16 + row) * 32 + 16 + lr] = __logf(fmaxf(cc11[r] * inv, 1e-37f)) + gscale;
  }
}

// ---------------------------------------------------------------------------
// Kernel B: scan the 65 composed matrices per batch -> boundary alpha/beta, logZ.
// ---------------------------------------------------------------------------
__global__ __launch_bounds__(32) void crf_boundary(float* __restrict__ ws,
                                                   float* __restrict__ out) {
  const int b = blockIdx.x, lane = threadIdx.x;
  float* Aw = ws + AW_OFF + (size_t)b * (NC + 1) * 32;
  float* Bw = ws + BW_OFF + (size_t)b * (NC + 1) * 32;
  float* LA = ws + LA_OFF + (size_t)b * NODES * 32;
  const float* Pf = ws + PF_OFF + (size_t)b * NC * 1024;

  float a = (lane == 0) ? 0.0f : NEGV;
  Aw[lane] = a;
  LA[lane] = a;
  for (int c = 0; c < NC; ++c) {
    const float* P = Pf + (size_t)c * 1024;
    float mx = -3.0e38f, sm = 0.0f;
    for (int i = 0; i < 32; ++i) {
      const float v = __shfl(a, i, 32) + P[i * 32 + lane];
      if (v > mx) { sm = sm * __expf(mx - v) + 1.0f; mx = v; }
      else        { sm += __expf(v - mx); }
    }
    a = mx + __logf(sm);
    Aw[(size_t)(c + 1) * 32 + lane] = a;
  }
  if (lane == 0) out[b] = a;  // logZ

  float bb = (lane == 0) ? 0.0f : NEGV;
  Bw[(size_t)NC * 32 + lane] = bb;
  for (int c = NC - 1; c >= 0; --c) {
    const float* P = Pf + (size_t)c * 1024;
    float mx = -3.0e38f, sm = 0.0f;
    for (int j = 0; j < 32; ++j) {
      const float v = P[lane * 32 + j] + __shfl(bb, j, 32);
      if (v > mx) { sm = sm * __expf(mx - v) + 1.0f; mx = v; }
      else        { sm += __expf(v - mx); }
    }
    bb = mx + __logf(sm);
    Bw[(size_t)c * 32 + lane] = bb;
  }
}

// ---------------------------------------------------------------------------
// Kernel C1: per-(b,chunk) wave replays forward alphas exactly, stores log_alpha.
// ---------------------------------------------------------------------------
__global__ __launch_bounds__(32) void crf_fwd_replay(const float* __restrict__ lm,
                                                     float* __restrict__ ws) {
  const int c = blockIdx.x, b = blockIdx.y, lane = threadIdx.x;
  const int L = min(CH, T1 - c * CH);
  float* LA = ws + LA_OFF + (size_t)b * NODES * 32;
  float a = ws[AW_OFF + ((size_t)b * (NC + 1) + c) * 32 + lane];

  for (int s = 0; s < L; ++s) {
    const size_t t = (size_t)c * CH + s;
    const float* mp = lm + ((size_t)b * T1 + t) * 1024;
    if (s + 1 < L) __builtin_prefetch(mp + 1024 + lane * 32, 0, 0);

    float mcol[32];
#pragma unroll
    for (int i = 0; i < 32; ++i) mcol[i] = mp[i * 32 + lane];

    float mx = -3.0e38f;
#pragma unroll
    for (int i = 0; i < 32; ++i) mx = fmaxf(mx, __shfl(a, i, 32) + mcol[i]);
    float sm = 0.0f;
#pragma unroll
    for (int i = 0; i < 32; ++i) sm += __expf(__shfl(a, i, 32) + mcol[i] - mx);
    a = mx + __logf(sm);
    LA[(t + 1) * 32 + lane] = a;
  }
}

// ---------------------------------------------------------------------------
// Kernel C2: per-(b,chunk) 256-thread block replays beta backward and emits
// unary + pair. m_t is double-buffered in LDS via async-to-LDS prefetch.
// ---------------------------------------------------------------------------
__global__ __launch_bounds__(256) void crf_bwd_outputs(const float* __restrict__ lm,
                                                       const float* __restrict__ ws,
                                                       float* __restrict__ out) {
  const int c = blockIdx.x, b = blockIdx.y, tid = threadIdx.x;
  const int L = min(CH, T1 - c * CH);
  const float* LA = ws + LA_OFF + (size_t)b * NODES * 32;
  float* unary = out + 32 + (size_t)b * NODES * 32;
  float* pair  = out + 32 + (size_t)B_ * NODES * 32 + (size_t)b * T1 * 1024;
  const float* mbase = lm + ((size_t)b * T1 + (size_t)c * CH) * 1024;

  __shared__ __align__(16) float mb[2][1024];
  __shared__ float bbuf[2][32];
  float* bc = bbuf[0];
  float* bn = bbuf[1];

  if (tid < 32) bc[tid] = ws[BW_OFF + ((size_t)b * (NC + 1) + c + 1) * 32 + tid];
  const float lZ = out[b];

  // prologue: async-load matrix for s=L-1 into buffer 0 (16B per lane, 8 waves)
  async_b128(lds_off(&mb[0][tid * 4]), (uint32_t)((L - 1) * 4096 + tid * 16), mbase);
  wait_async0();
  __syncthreads();

  const int i4 = tid >> 3;
  const int j4 = (tid & 7) * 4;

  for (int s = L - 1; s >= 0; --s) {
    const int cur = (L - 1 - s) & 1;
    if (s > 0)  // prefetch next matrix into the other buffer
      async_b128(lds_off(&mb[1 - cur][tid * 4]),
                 (uint32_t)((s - 1) * 4096 + tid * 16), mbase);

    const size_t t = (size_t)c * CH + s;

    // unary at node t+1 uses beta(t+1) = bc
    if (tid < 32)
      unary[(t + 1) * 32 + tid] = __expf(LA[(t + 1) * 32 + tid] + bc[tid] - lZ);

    // pair[t][i][j] = exp(la_t[i] + m[i][j] + beta_{t+1}[j] - logZ)
    const float la_ti = LA[t * 32 + i4];
    const float4 mv = ((const float4*)mb[cur])[tid];
    float4 pv;
    pv.x = __expf(la_ti + mv.x + bc[j4 + 0] - lZ);
    pv.y = __expf(la_ti + mv.y + bc[j4 + 1] - lZ);
    pv.z = __expf(la_ti + mv.z + bc[j4 + 2] - lZ);
    pv.w = __expf(la_ti + mv.w + bc[j4 + 3] - lZ);
    ((float4*)(pair + t * 1024))[tid] = pv;
    __syncthreads();

    // beta step: wave 0, lane i: beta_t[i] = logsumexp_j(m[i][j] + beta_{t+1}[j])
    if (tid < 32) {
      float mx = -3.0e38f, sm = 0.0f;
#pragma unroll
      for (int j = 0; j < 32; ++j) {
        const float v = mb[cur][tid * 32 + j] + bc[j];
        if (v > mx) { sm = sm * __expf(mx - v) + 1.0f; mx = v; }
        else        { sm += __expf(v - mx); }
      }
      bn[tid] = mx + __logf(sm);
    }
    __syncthreads();
    float* tmp = bc; bc = bn; bn = tmp;

    if (s > 0) {         // next buffer fully landed & visible to all waves
      wait_async0();
      __syncthreads();
    }
  }

  if (c == 0 && tid < 32)
    unary[tid] = __expf(LA[tid] + bc[tid] - lZ);
}

// ---------------------------------------------------------------------------
extern "C" void kernel_launch(void* const* d_in, const int* in_sizes, int n_in,
                              void* d_out, int out_size, void* d_ws, size_t ws_size,
                              hipStream_t stream) {
  const float* lm = (const float*)d_in[0];
  float* out = (float*)d_out;
  float* ws  = (float*)d_ws;   // needs ~25.8 MB

  const dim3 gChunks(NC, B_);
  crf_compose    <<<gChunks, dim3(32),  0, stream>>>(lm, ws);
  crf_boundary   <<<dim3(B_), dim3(32), 0, stream>>>(ws, out);
  crf_fwd_replay <<<gChunks, dim3(32),  0, stream>>>(lm, ws);
  crf_bwd_outputs<<<gChunks, dim3(256), 0, stream>>>(lm, ws, out);
}